// Bert_BiLSTM_CRF_31533649887466
// MI455X (gfx1250) — compile-verified
//
#include <hip/hip_runtime.h>
#include <hip/hip_bf16.h>

typedef __attribute__((ext_vector_type(2))) float v2f;
typedef __attribute__((ext_vector_type(8))) float v8f;

#define BB 64
#define TT 512
#define HH 768
#define KK 11
#define KP 16            // K padded to 16 for WMMA tile
#define START_ID 9
#define NEGV (-10000.0f)

// ---------------------------------------------------------------------------
// Kernel 1: feats = embeds([M=B*T, H]) @ W_fc([K,H])^T + b_fc, via fp32 WMMA.
// One wave (32 threads) per 16-row tile; 192 chained V_WMMA_F32_16X16X4_F32.
// Stores padded [M, 16] tile (cols >= 11 are junk, never read back).
// ---------------------------------------------------------------------------
__global__ __launch_bounds__(32) void fc_wmma_f32_kernel(
    const float* __restrict__ A,      // embeds, [M, H]
    const float* __restrict__ W,      // W_fc,   [K, H]
    const float* __restrict__ bias,   // b_fc,   [K]
    float* __restrict__ featsP)       // out,    [M, KP]
{
    const int m0   = blockIdx.x * 16;
    const int lane = threadIdx.x;      // 0..31, wave32
    const int half = lane >> 4;        // 0: K={k0,k0+1}; 1: K={k0+2,k0+3}
    const int l16  = lane & 15;

    // A fragment: lane l16 owns row M=l16; VGPR0/1 hold K = k0 + 2*half + {0,1}
    const float* arow = A + (size_t)(m0 + l16) * HH + 2 * half;
    // B fragment: lane l16 owns column N=l16 -> W row l16 (clamped for pad cols)
    const int wcol = (l16 < KK) ? l16 : (KK - 1);
    const float* wrow = W + (size_t)wcol * HH + 2 * half;

    v8f c = {};
    #pragma unroll 8
    for (int k0 = 0; k0 < HH; k0 += 4) {
        v2f a, b;
        a.x = arow[k0];  a.y = arow[k0 + 1];
        b.x = wrow[k0];  b.y = wrow[k0 + 1];
        // D = A(16x4,f32) * B(4x16,f32) + C(16x16,f32)
        c = __builtin_amdgcn_wmma_f32_16x16x4_f32(
                /*neg_a=*/false, a, /*neg_b=*/false, b,
                /*c_mod=*/(short)0, c, /*reuse_a=*/false, /*reuse_b=*/false);
    }

    const float bcol = bias[wcol];
    // C/D layout: VGPR r -> row M = r + 8*half (lanes 0-15: M=r, 16-31: M=r+8),
    // column N = l16.
    #pragma unroll
    for (int r = 0; r < 8; ++r) {
        const int m = m0 + r + 8 * half;
        featsP[(size_t)m * KP + l16] = c[r] + bcol;
    }
}

// ---------------------------------------------------------------------------
// Kernel 2: Viterbi forward scan + backtrack. One wave per batch element.
// Lane i (< 11) owns state i; log_delta broadcast via wave32 shuffles.
// psi stored in LDS (512 x 16 ints = 32 KB of the 320 KB WGP pool).
// ---------------------------------------------------------------------------
__global__ __launch_bounds__(32) void viterbi_kernel(
    const float* __restrict__ featsP,   // [B, T, KP]
    const float* __restrict__ trans,    // [K, K] row i = dest, col j = src
    float* __restrict__ score_out,      // [B]
    float* __restrict__ path_out)       // [B, T] (tags as floats)
{
    __shared__ int psi[TT][KP];

    const int b    = blockIdx.x;
    const int lane = threadIdx.x;      // state id for lanes < KK

    // Transition row for my destination state, kept in registers.
    float tr[KK];
    #pragma unroll
    for (int j = 0; j < KK; ++j)
        tr[j] = (lane < KK) ? trans[lane * KK + j] : NEGV;

    // log_delta0: NEG everywhere except START tag (feats[:,0] intentionally
    // skipped, matching the reference).
    float delta = (lane == START_ID) ? 0.0f : NEGV;

    const float* fb = featsP + (size_t)b * TT * KP;

    for (int t = 1; t < TT; ++t) {
        if (t + 8 < TT)
            __builtin_prefetch(&fb[(size_t)(t + 8) * KP + (lane & (KP - 1))], 0, 1);

        float best = -3.0e38f;
        int   arg  = 0;
        #pragma unroll
        for (int j = 0; j < KK; ++j) {
            const float dj = __shfl(delta, j, 32);   // broadcast log_delta[j]
            const float s  = tr[j] + dj;
            if (s > best) { best = s; arg = j; }     // strict '>' == first argmax
        }
        const float ft = (lane < KK) ? fb[(size_t)t * KP + lane] : 0.0f;
        delta = best + ft;
        if (lane < KK) psi[t][lane] = arg;
    }

    // Final reduce: score = max_i delta_i, last_tag = first argmax.
    float m = -3.0e38f;
    int argi = 0;
    #pragma unroll
    for (int j = 0; j < KK; ++j) {
        const float dj = __shfl(delta, j, 32);
        if (dj > m) { m = dj; argi = j; }
    }

    __syncthreads();   // single wave: just orders LDS writes before backtrack

    if (lane == 0) {
        score_out[b] = m;
        float* pb = path_out + (size_t)b * TT;
        int nxt = argi;
        pb[TT - 1] = (float)nxt;
        for (int t = TT - 2; t >= 0; --t) {
            nxt = psi[t + 1][nxt];
            pb[t] = (float)nxt;
        }
    }
}

// ---------------------------------------------------------------------------
extern "C" void kernel_launch(void* const* d_in, const int* in_sizes, int n_in,
                              void* d_out, int out_size, void* d_ws, size_t ws_size,
                              hipStream_t stream) {
    (void)in_sizes; (void)n_in; (void)out_size; (void)ws_size;

    const float* embeds = (const float*)d_in[0];   // [B,T,H]
    const float* W_fc   = (const float*)d_in[1];   // [K,H]
    const float* b_fc   = (const float*)d_in[2];   // [K]
    const float* trans  = (const float*)d_in[3];   // [K,K]

    float* out      = (float*)d_out;
    float* score    = out;                // [B]
    float* path     = out + BB;           // [B,T]
    float* featsP   = (float*)d_ws;       // [B*T, KP] = 2 MB scratch

    const int M = BB * TT;                // 32768 rows
    fc_wmma_f32_kernel<<<M / 16, 32, 0, stream>>>(embeds, W_fc, b_fc, featsP);
    viterbi_kernel<<<BB, 32, 0, stream>>>(featsP, trans, score, path);
}